// batch_pearson_loss_wave_15444702397065
// MI455X (gfx1250) — compile-verified
//
#include <hip/hip_runtime.h>
#include <math.h>

typedef float v2f __attribute__((ext_vector_type(2)));
typedef float v4f __attribute__((ext_vector_type(4)));
typedef float v8f __attribute__((ext_vector_type(8)));

#define ROW_LEN 4096
#define BETA_F 1e-7f

// Full 32-lane sum using V_WMMA_F32_16X16X4_F32.
// A[16x4]: lane L<16 holds A[L,0]=partial (vgpr0), A[L,1]=0 (vgpr1);
//          lane L>=16 holds A[L-16,2]=partial, A[L-16,3]=0.
// B[4x16] = all ones  =>  D[m,n] = partial[m] + partial[m+16].
// Lane sums its 8 D vgprs (rows m=0..7 or 8..15 of its column), then one
// xor-16 shuffle combines the two halves -> grand total in every lane.
__device__ __forceinline__ float wave_reduce_sum_wmma(float partial) {
    v2f a; a[0] = partial; a[1] = 0.0f;
    v2f b; b[0] = 1.0f;    b[1] = 1.0f;
    v8f c = {};
    c = __builtin_amdgcn_wmma_f32_16x16x4_f32(
        /*neg_a=*/false, a, /*neg_b=*/false, b,
        /*c_mod=*/(short)0, c, /*reuse_a=*/false, /*reuse_b=*/false);
    float s = ((c[0] + c[1]) + (c[2] + c[3])) + ((c[4] + c[5]) + (c[6] + c[7]));
    s += __shfl_xor(s, 16, 32);
    return s;
}

// One wave32 per row: streaming NT b128 loads, 5 FMA accumulators per lane,
// WMMA-based cross-lane reduction, lane 0 writes per-row loss.
__global__ void __launch_bounds__(256)
pearson_rows_kernel(const float* __restrict__ x,
                    const float* __restrict__ y,
                    float* __restrict__ p_rows,
                    int n_rows) {
    const int wave = (int)((blockIdx.x * (unsigned)blockDim.x + threadIdx.x) >> 5);
    const int lane = (int)(threadIdx.x & 31u);
    if (wave >= n_rows) return;          // wave-uniform: EXEC stays all-1s below

    const float* __restrict__ xr = x + (long long)wave * ROW_LEN;
    const float* __restrict__ yr = y + (long long)wave * ROW_LEN;

    float sx = 0.0f, sy = 0.0f, sxy = 0.0f, sxx = 0.0f, syy = 0.0f;

    // 32 lanes x float4 = 128 elements per pass; 32 passes per 4096-elem row.
    #pragma unroll 4
    for (int pass = 0; pass < ROW_LEN / 128; ++pass) {
        const int col = pass * 128 + lane * 4;
        v4f xv = __builtin_nontemporal_load((const v4f*)(xr + col));
        v4f yv = __builtin_nontemporal_load((const v4f*)(yr + col));
        #pragma unroll
        for (int j = 0; j < 4; ++j) {
            const float xe = xv[j], ye = yv[j];
            sx  += xe;
            sy  += ye;
            sxy = fmaf(xe, ye, sxy);
            sxx = fmaf(xe, xe, sxx);
            syy = fmaf(ye, ye, syy);
        }
    }

    // Cross-lane reductions through the matrix pipe (v_wmma_f32_16x16x4_f32).
    sx  = wave_reduce_sum_wmma(sx);
    sy  = wave_reduce_sum_wmma(sy);
    sxy = wave_reduce_sum_wmma(sxy);
    sxx = wave_reduce_sum_wmma(sxx);
    syy = wave_reduce_sum_wmma(syy);

    if (lane == 0) {
        const float invT = 1.0f / (float)ROW_LEN;
        const float num = sxy - sx * sy * invT;
        const float vx  = sxx - sx * sx * invT;
        const float vy  = syy - sy * sy * invT;
        const float den = sqrtf(vx + BETA_F) * sqrtf(vy + BETA_F);
        p_rows[wave] = 1.0f - num / den;
    }
}

// Deterministic single-block mean over the per-row losses (fixed-order tree).
__global__ void __launch_bounds__(256)
mean_reduce_kernel(const float* __restrict__ p_rows,
                   float* __restrict__ out,
                   int n_rows) {
    __shared__ float smem[256];
    float s = 0.0f;
    for (int i = (int)threadIdx.x; i < n_rows; i += 256) s += p_rows[i];
    smem[threadIdx.x] = s;
    __syncthreads();
    #pragma unroll
    for (int off = 128; off > 0; off >>= 1) {
        if ((int)threadIdx.x < off) smem[threadIdx.x] += smem[threadIdx.x + off];
        __syncthreads();
    }
    if (threadIdx.x == 0) out[0] = smem[0] * (1.0f / (float)n_rows);
}

extern "C" void kernel_launch(void* const* d_in, const int* in_sizes, int n_in,
                              void* d_out, int out_size, void* d_ws, size_t ws_size,
                              hipStream_t stream) {
    const float* predict = (const float*)d_in[0];
    const float* target  = (const float*)d_in[1];
    float* p_rows = (float*)d_ws;          // n_rows floats of scratch
    float* out    = (float*)d_out;         // single fp32 scalar

    const int n_rows = in_sizes[0] / ROW_LEN;   // 8192 for the reference shapes

    const int threads = 256;                    // 8 wave32s per block
    const int blocks  = (n_rows * 32 + threads - 1) / threads;

    pearson_rows_kernel<<<blocks, threads, 0, stream>>>(predict, target, p_rows, n_rows);
    mean_reduce_kernel<<<1, 256, 0, stream>>>(p_rows, out, n_rows);
}